// HashEncoder_49220325212769
// MI455X (gfx1250) — compile-verified
//
#include <hip/hip_runtime.h>
#include <math.h>

// -----------------------------------------------------------------------------
// Instant-NGP style hash-grid encoder for MI455X (gfx1250, wave32).
//
// Gather-bound kernel; 128MB of hash tables fit the 192MB L2, so gathers are
// L2-resident. Output (256MB) is written with non-temporal (TH=NT) stores so
// the streaming writes do not evict the tables. Positions are staged coalesced
// into LDS via the gfx1250 async-to-LDS path (ASYNCcnt + s_wait_asynccnt).
// -----------------------------------------------------------------------------

namespace {
constexpr int      kLevels    = 16;
constexpr unsigned kTableSize = 1u << 20;
constexpr unsigned kHashMask  = kTableSize - 1u;
constexpr unsigned kP1        = 2654435761u;
constexpr unsigned kP2        = 805459861u;
constexpr int      kBlock     = 256;   // 8 waves (wave32) per block

struct ResArr { int r[kLevels]; };
} // namespace

typedef __attribute__((address_space(1))) int g_i32;  // global
typedef __attribute__((address_space(3))) int l_i32;  // LDS

__device__ __forceinline__ void wait_asynccnt0() {
#if __has_builtin(__builtin_amdgcn_s_wait_asynccnt)
  __builtin_amdgcn_s_wait_asynccnt(0);
#else
  asm volatile("s_wait_asynccnt 0" ::: "memory");
#endif
}

__device__ __forceinline__ float clamp01(float v) {
  return fminf(fmaxf(v, 0.0f), 1.0f);
}

__global__ __launch_bounds__(kBlock) void hashgrid_encode_kernel(
    const float*  __restrict__ pos,      // [N,3]
    const float2* __restrict__ tables,   // [16, 2^20] float2
    float2*       __restrict__ out,      // [N,16] float2  (== [N,32] f32)
    int n_pts, ResArr res)
{
  __shared__ float sp[kBlock * 3];
  const int tid = threadIdx.x;
  const long long base3 = (long long)blockIdx.x * (long long)(kBlock * 3);
  const long long gmax  = (long long)n_pts * 3 - 1;

  // --- Stage this block's 768 position floats into LDS, coalesced. ---------
#pragma unroll
  for (int s = 0; s < 3; ++s) {
    long long gi = base3 + tid + s * kBlock;
    if (gi > gmax) gi = gmax;               // clamp: tail lanes read last elem
    const int li = tid + s * kBlock;
#if __has_builtin(__builtin_amdgcn_global_load_async_to_lds_b32)
    // gfx1250 async DMA: global -> LDS, tracked by ASYNCcnt.
    __builtin_amdgcn_global_load_async_to_lds_b32(
        (g_i32*)(pos + gi),
        (l_i32*)&sp[li],
        0 /*imm offset*/, 0 /*cpol*/);
#else
    sp[li] = pos[gi];
#endif
  }
#if __has_builtin(__builtin_amdgcn_global_load_async_to_lds_b32)
  wait_asynccnt0();
#endif
  __syncthreads();

  const int n = blockIdx.x * kBlock + tid;
  // LDS read: addresses 3t,3t+1,3t+2 — gcd(3,64)=1 => bank-conflict-free.
  float px = sp[tid * 3 + 0];
  float py = sp[tid * 3 + 1];
  float pz = sp[tid * 3 + 2];
  if (n >= n_pts) return;

  px = clamp01(px); py = clamp01(py); pz = clamp01(pz);

  float2* __restrict__ op = out + (size_t)n * kLevels;

#pragma unroll
  for (int l = 0; l < kLevels; ++l) {       // full unroll: res.r[l] constant-indexed
    const int   rl = res.r[l];
    const float rf = (float)rl;

    const float sx = px * rf, sy = py * rf, sz = pz * rf;
    const float fx = floorf(sx), fy = floorf(sy), fz = floorf(sz);
    const float tx = sx - fx,   ty = sy - fy,   tz = sz - fz;

    int cx = (int)fx, cy = (int)fy, cz = (int)fz;
    const int rm = rl - 1;
    cx = cx < 0 ? 0 : (cx > rm ? rm : cx);
    cy = cy < 0 ? 0 : (cy > rm ? rm : cy);
    cz = cz < 0 ? 0 : (cz > rm ? rm : cz);

    // Incremental hashing: corner (i,j,k) = h000 + i*1 + j*P1 + k*P2 (mod 2^32),
    // then mask (TABLE_SIZE is a power of two). Matches uint32 wraparound of ref.
    const unsigned h = (unsigned)cx + (unsigned)cy * kP1 + (unsigned)cz * kP2;
    const float2* __restrict__ t = tables + (size_t)l * kTableSize;

    // 8 independent b64 gathers -> deep MLP against L2.
    const float2 f000 = t[(h)                  & kHashMask];
    const float2 f001 = t[(h + kP2)            & kHashMask];
    const float2 f010 = t[(h + kP1)            & kHashMask];
    const float2 f011 = t[(h + kP1 + kP2)      & kHashMask];
    const float2 f100 = t[(h + 1u)             & kHashMask];
    const float2 f101 = t[(h + 1u + kP2)       & kHashMask];
    const float2 f110 = t[(h + 1u + kP1)       & kHashMask];
    const float2 f111 = t[(h + 1u + kP1 + kP2) & kHashMask];

    // Replicate the reference's lerp structure exactly (tx blends the k-axis
    // pairs, as written in the reference).
    const float ox = 1.0f - tx, oy = 1.0f - ty, oz = 1.0f - tz;
    const float c00x = f000.x * ox + f001.x * tx, c00y = f000.y * ox + f001.y * tx;
    const float c01x = f010.x * ox + f011.x * tx, c01y = f010.y * ox + f011.y * tx;
    const float c10x = f100.x * ox + f101.x * tx, c10y = f100.y * ox + f101.y * tx;
    const float c11x = f110.x * ox + f111.x * tx, c11y = f110.y * ox + f111.y * tx;
    const float c0x  = c00x * oy + c10x * ty,     c0y  = c00y * oy + c10y * ty;
    const float c1x  = c01x * oy + c11x * ty,     c1y  = c01y * oy + c11y * ty;
    const float rx   = c0x * oz + c1x * tz,       ry   = c0y * oz + c1y * tz;

    // Non-temporal b64 store: keep the 256MB output stream from evicting the
    // L2-resident hash tables. Each point's 32 floats = exactly one 128B line.
    union { float2 f; unsigned long long u; } cv;
    cv.f = make_float2(rx, ry);
    __builtin_nontemporal_store(cv.u,
        reinterpret_cast<unsigned long long*>(op + l));
  }
}

extern "C" void kernel_launch(void* const* d_in, const int* in_sizes, int n_in,
                              void* d_out, int out_size, void* d_ws, size_t ws_size,
                              hipStream_t stream) {
  (void)n_in; (void)out_size; (void)d_ws; (void)ws_size;
  const float*  pos    = (const float*)d_in[0];
  const float2* tables = (const float2*)d_in[1];
  float2*       out    = (float2*)d_out;
  const int n_pts = in_sizes[0] / 3;

  // Match the reference's float64 resolution computation bit-for-bit
  // (int() truncation included) instead of hand-baking constants.
  ResArr res;
  const double b = exp(log(2048.0 / 16.0) / 15.0);
  for (int l = 0; l < kLevels; ++l)
    res.r[l] = (int)(16.0 * pow(b, (double)l));

  const int grid = (n_pts + kBlock - 1) / kBlock;
  hashgrid_encode_kernel<<<grid, kBlock, 0, stream>>>(pos, tables, out, n_pts, res);
}